// DynamicLinBertSelfAttention_43026982371726
// MI455X (gfx1250) — compile-verified
//
#include <hip/hip_runtime.h>
#include <math.h>

typedef __attribute__((ext_vector_type(2))) float v2f;
typedef __attribute__((ext_vector_type(8))) float v8f;

#define BSZ   4
#define SEQ   2048
#define HDIM  1024
#define NH    16
#define HD    64
#define RMAX  64
#define MTOT  (BSZ*SEQ)   /* 8192 */

__device__ __forceinline__ v8f wmma4(v2f a, v2f b, v8f c) {
  // V_WMMA_F32_16X16X4_F32 : D = A(16x4 f32) * B(4x16 f32) + C(16x16 f32)
  return __builtin_amdgcn_wmma_f32_16x16x4_f32(
      /*neg_a=*/false, a, /*neg_b=*/false, b,
      /*c_mod=*/(short)0, c, /*reuse_a=*/false, /*reuse_b=*/false);
}

// ---------------------------------------------------------------------------
// C[M,N] = A[M,K] @ W[N,K]^T + bias[N]   (M=8192, N=K=1024 here)
// block = 256 threads (8 waves); C tile 128x64; K tile 32 staged in LDS.
// ---------------------------------------------------------------------------
__global__ void __launch_bounds__(256)
gemm_xwt_kernel(const float* __restrict__ A, const float* __restrict__ W,
                const float* __restrict__ bias, float* __restrict__ C,
                int M, int N, int K) {
  __shared__ __align__(16) float As[128][36];
  __shared__ __align__(16) float Ws[64][36];
  const int tid  = threadIdx.x;
  const int wave = tid >> 5;
  const int lane = tid & 31;
  const int l16  = lane & 15;
  const int half = lane >> 4;
  const int m0 = blockIdx.y * 128;
  const int n0 = blockIdx.x * 64;

  v8f acc[4] = {};

  for (int k0 = 0; k0 < K; k0 += 32) {
    // ---- stage tiles (coalesced float4) ----
    {
      const int f = tid & 7;     // float4 column within 32-wide tile
      const int r = tid >> 3;    // 0..31
#pragma unroll
      for (int rr = 0; rr < 4; ++rr) {
        const int row = r + rr * 32;
        *(float4*)&As[row][f * 4] =
            *(const float4*)(A + (size_t)(m0 + row) * K + k0 + f * 4);
      }
#pragma unroll
      for (int rr = 0; rr < 2; ++rr) {
        const int row = r + rr * 32;
        *(float4*)&Ws[row][f * 4] =
            *(const float4*)(W + (size_t)(n0 + row) * K + k0 + f * 4);
      }
    }
    __syncthreads();

    const int am = wave * 16 + l16;
#pragma unroll
    for (int kk = 0; kk < 8; ++kk) {
      const int kb = kk * 4 + 2 * half;               // K = kb, kb+1 per lane-half
      v2f afrag = *(const v2f*)&As[am][kb];
#pragma unroll
      for (int nt = 0; nt < 4; ++nt) {
        v2f bfrag = *(const v2f*)&Ws[nt * 16 + l16][kb];  // B[k][n] = W[n][k]
        acc[nt] = wmma4(afrag, bfrag, acc[nt]);
      }
    }
    __syncthreads();
  }

#pragma unroll
  for (int nt = 0; nt < 4; ++nt) {
    const int col = n0 + nt * 16 + l16;
    const float bv = bias[col];
#pragma unroll
    for (int r = 0; r < 8; ++r) {
      const int row = m0 + wave * 16 + r + 8 * half;  // C: VGPR r -> M=r / r+8
      C[(size_t)row * N + col] = acc[nt][r] + bv;
    }
  }
}

// ---------------------------------------------------------------------------
// kp[b,h,r,d] = sum_l proj[h,l,r] * KV[(b*S+l), h*HD+d]
// grid.x = b*NH+h (64), grid.y = 0..1 ; 8 waves, one 16x16 tile each.
// ---------------------------------------------------------------------------
__global__ void __launch_bounds__(256)
proj_kernel(const float* __restrict__ KV, const float* __restrict__ P,
            float* __restrict__ OUT) {
  const int bh = blockIdx.x;
  const int b = bh >> 4, h = bh & 15;
  const int wave = threadIdx.x >> 5;
  const int lane = threadIdx.x & 31;
  const int l16 = lane & 15, half = lane >> 4;
  const int tile = blockIdx.y * 8 + wave;   // 0..15
  const int rt = tile >> 2, dt = tile & 3;

  const float* Ph = P + (size_t)h * SEQ * RMAX;
  const float* Kb = KV + (size_t)b * SEQ * HDIM + h * HD;

  v8f acc = {};
  for (int l0 = 0; l0 < SEQ; l0 += 4) {
    const int lA = l0 + 2 * half;
    v2f a, bf;
    a.x  = Ph[(size_t)lA * RMAX + rt * 16 + l16];        // A[r][l] = proj[l][r]
    a.y  = Ph[(size_t)(lA + 1) * RMAX + rt * 16 + l16];
    bf.x = Kb[(size_t)lA * HDIM + dt * 16 + l16];        // B[l][d] = k[l][d]
    bf.y = Kb[(size_t)(lA + 1) * HDIM + dt * 16 + l16];
    acc = wmma4(a, bf, acc);
  }

  float* Ob = OUT + (size_t)bh * RMAX * HD;
#pragma unroll
  for (int r = 0; r < 8; ++r) {
    const int rr = rt * 16 + r + 8 * half;
    Ob[rr * HD + dt * 16 + l16] = acc[r];
  }
}

// ---------------------------------------------------------------------------
// Fused: scores = Q@kp^T/8 -> mask(r>=curr_r) -> softmax -> attn out
//        context = P @ vp     -> CTX
// grid: (SEQ/128, BSZ*NH); 8 waves, each owns 16 s-rows.
// ---------------------------------------------------------------------------
__global__ void __launch_bounds__(256)
attn_kernel(const float* __restrict__ Q, const float* __restrict__ KP,
            const float* __restrict__ VP, const int* __restrict__ currR,
            float* __restrict__ ATTN, float* __restrict__ CTX) {
  __shared__ __align__(16) float kps[64][66];       // kp[r][d]
  __shared__ __align__(16) float vpt[64][66];       // vp^T : vpt[d][r]
  __shared__ __align__(16) float pls[8][16][66];    // per-wave prob tile

  const int bh = blockIdx.y;
  const int b = bh >> 4, h = bh & 15;
  const int s0 = blockIdx.x * 128;
  const int tid = threadIdx.x;
  const int wave = tid >> 5, lane = tid & 31;
  const int l16 = lane & 15, half = lane >> 4;
  const int r_valid = *currR;

  {
    const int c = tid & 63;
    const int rb = tid >> 6;            // 0..3
    const float* kpb = KP + (size_t)bh * RMAX * HD;
    const float* vpb = VP + (size_t)bh * RMAX * HD;
#pragma unroll
    for (int rr = 0; rr < 16; ++rr) {
      const int row = rb + rr * 4;
      kps[row][c] = kpb[row * HD + c];
      vpt[c][row] = vpb[row * HD + c];
    }
  }
  __syncthreads();

  // ---- scores: A[s][d] = Q, B[d][r] = kp[r][d] ----
  v8f sc[4] = {};
  const float* Qb = Q + ((size_t)(b * SEQ + s0 + wave * 16 + l16)) * HDIM + h * HD;
#pragma unroll
  for (int kk = 0; kk < 16; ++kk) {
    const int kb = kk * 4 + 2 * half;
    v2f a = *(const v2f*)(Qb + kb);
#pragma unroll
    for (int nt = 0; nt < 4; ++nt) {
      v2f bf = *(const v2f*)&kps[nt * 16 + l16][kb];
      sc[nt] = wmma4(a, bf, sc[nt]);
    }
  }

  // ---- masked softmax over r (row lives in 16 lanes of this half x 4 accs) ----
  const float scale = 0.125f;  // 1/sqrt(64)
#pragma unroll
  for (int rv = 0; rv < 8; ++rv) {
    float v0[4];
    float mx = -INFINITY;
#pragma unroll
    for (int nt = 0; nt < 4; ++nt) {
      const int r = nt * 16 + l16;
      float s = sc[nt][rv] * scale;
      s = (r < r_valid) ? s : -INFINITY;
      v0[nt] = s;
      mx = fmaxf(mx, s);
    }
    for (int m = 1; m < 16; m <<= 1) mx = fmaxf(mx, __shfl_xor(mx, m, 32));
    float sum = 0.f;
#pragma unroll
    for (int nt = 0; nt < 4; ++nt) { v0[nt] = expf(v0[nt] - mx); sum += v0[nt]; }
    for (int m = 1; m < 16; m <<= 1) sum += __shfl_xor(sum, m, 32);
    const float inv = 1.0f / sum;

    const int srow = s0 + wave * 16 + rv + 8 * half;
    const size_t abase = ((size_t)bh * SEQ + srow) * RMAX;
#pragma unroll
    for (int nt = 0; nt < 4; ++nt) {
      const float p = v0[nt] * inv;
      ATTN[abase + nt * 16 + l16] = p;
      pls[wave][rv + 8 * half][nt * 16 + l16] = p;   // per-wave, in-order LDS
    }
  }

  // ---- context: A[s][r] = P (LDS), B[r][d] = vp[r][d] = vpt[d][r] ----
  v8f cacc[4] = {};
#pragma unroll
  for (int kk = 0; kk < 16; ++kk) {
    const int kb = kk * 4 + 2 * half;
    v2f a = *(const v2f*)&pls[wave][l16][kb];
#pragma unroll
    for (int nt = 0; nt < 4; ++nt) {
      v2f bf = *(const v2f*)&vpt[nt * 16 + l16][kb];
      cacc[nt] = wmma4(a, bf, cacc[nt]);
    }
  }
#pragma unroll
  for (int nt = 0; nt < 4; ++nt) {
    const int dd = nt * 16 + l16;
#pragma unroll
    for (int r = 0; r < 8; ++r) {
      const int srow = s0 + wave * 16 + r + 8 * half;
      CTX[((size_t)(b * SEQ + srow)) * HDIM + h * HD + dd] = cacc[nt][r];
    }
  }
}

// ---------------------------------------------------------------------------
// per-sample sum / sumsq (double), then curr_r
// ---------------------------------------------------------------------------
__global__ void __launch_bounds__(256)
var_kernel(const float* __restrict__ X, double* __restrict__ stats) {
  __shared__ double ssum[256], ssq[256];
  const int b = blockIdx.x;
  const float* xb = X + (size_t)b * SEQ * HDIM;
  double s = 0.0, q = 0.0;
  for (int i = threadIdx.x; i < SEQ * HDIM; i += 256) {
    const double x = (double)xb[i];
    s += x; q += x * x;
  }
  ssum[threadIdx.x] = s; ssq[threadIdx.x] = q;
  __syncthreads();
  for (int ofs = 128; ofs > 0; ofs >>= 1) {
    if (threadIdx.x < ofs) {
      ssum[threadIdx.x] += ssum[threadIdx.x + ofs];
      ssq[threadIdx.x]  += ssq[threadIdx.x + ofs];
    }
    __syncthreads();
  }
  if (threadIdx.x == 0) { stats[b * 2] = ssum[0]; stats[b * 2 + 1] = ssq[0]; }
}

__global__ void rank_kernel(const double* __restrict__ stats,
                            int* __restrict__ rOut, float* __restrict__ rFloatOut) {
  const double N = (double)(SEQ * HDIM);
  double vv[4], vmin = 1e300, vmax = -1e300;
  for (int b = 0; b < 4; ++b) {
    const double s = stats[2 * b], q = stats[2 * b + 1];
    const double var = (q - s * s / N) / (N - 1.0);   // ddof=1
    vv[b] = var;
    vmin = fmin(vmin, var); vmax = fmax(vmax, var);
  }
  double mean = 0.0;
  for (int b = 0; b < 4; ++b) mean += (vv[b] - vmin) / (vmax - vmin + 1e-6);
  mean *= 0.25;
  int r = (int)rint(16.0 + mean * 48.0);              // RNE like jnp.round
  if (r < 16) r = 16;
  if (r > 64) r = 64;
  *rOut = r;
  *rFloatOut = (float)r;
}

// ---------------------------------------------------------------------------
extern "C" void kernel_launch(void* const* d_in, const int* in_sizes, int n_in,
                              void* d_out, int out_size, void* d_ws, size_t ws_size,
                              hipStream_t stream) {
  (void)in_sizes; (void)n_in; (void)out_size; (void)ws_size;
  const float* hidden = (const float*)d_in[0];
  const float* Wq = (const float*)d_in[1];
  const float* bq = (const float*)d_in[2];
  const float* Wk = (const float*)d_in[3];
  const float* bk = (const float*)d_in[4];
  const float* Wv = (const float*)d_in[5];
  const float* bv = (const float*)d_in[6];
  const float* Wo = (const float*)d_in[7];
  const float* bo = (const float*)d_in[8];
  const float* projk = (const float*)d_in[9];
  const float* projv = (const float*)d_in[10];

  float* out  = (float*)d_out;                       // [4,2048,1024]
  float* attn = out + (size_t)MTOT * HDIM;           // [4,16,2048,64]
  float* rflt = attn + (size_t)BSZ * NH * SEQ * RMAX;// scalar

  char* ws = (char*)d_ws;
  float* q    = (float*)ws; ws += (size_t)MTOT * HDIM * 4;
  float* k    = (float*)ws; ws += (size_t)MTOT * HDIM * 4;
  float* vctx = (float*)ws; ws += (size_t)MTOT * HDIM * 4;   // v, later context
  float* kp   = (float*)ws; ws += (size_t)BSZ * NH * RMAX * HD * 4;
  float* vp   = (float*)ws; ws += (size_t)BSZ * NH * RMAX * HD * 4;
  double* stats = (double*)ws; ws += 16 * 8;
  int* rdev = (int*)ws;

  var_kernel<<<4, 256, 0, stream>>>(hidden, stats);
  rank_kernel<<<1, 1, 0, stream>>>(stats, rdev, rflt);

  dim3 g1(HDIM / 64, MTOT / 128);
  gemm_xwt_kernel<<<g1, 256, 0, stream>>>(hidden, Wq, bq, q,    MTOT, HDIM, HDIM);
  gemm_xwt_kernel<<<g1, 256, 0, stream>>>(hidden, Wk, bk, k,    MTOT, HDIM, HDIM);
  gemm_xwt_kernel<<<g1, 256, 0, stream>>>(hidden, Wv, bv, vctx, MTOT, HDIM, HDIM);

  proj_kernel<<<dim3(BSZ * NH, 2), 256, 0, stream>>>(k,    projk, kp);
  proj_kernel<<<dim3(BSZ * NH, 2), 256, 0, stream>>>(vctx, projv, vp);

  attn_kernel<<<dim3(SEQ / 128, BSZ * NH), 256, 0, stream>>>(q, kp, vp, rdev,
                                                             attn, vctx);

  gemm_xwt_kernel<<<g1, 256, 0, stream>>>(vctx, Wo, bo, out, MTOT, HDIM, HDIM);
}